// APModel_16355235463647
// MI455X (gfx1250) — compile-verified
//
#include <hip/hip_runtime.h>

typedef float v2f __attribute__((ext_vector_type(2)));
typedef float v8f __attribute__((ext_vector_type(8)));

#define DIM_D 8192
#define WAVES_PER_BLOCK 8
#define K_PER_WAVE (DIM_D / WAVES_PER_BLOCK) /* 1024 */

// One block per 16-row tile of S. Each of the 8 wave32s covers a 1024-wide
// K-slice with V_WMMA_F32_16X16X4_F32 (matvec packed into column 0 of B),
// then partial row-sums are reduced across waves through LDS and the
// element-wise reaction terms are fused into the final write.
//
// HBM roofline: S = 256 MB streamed once @ 23.3 TB/s ~= 11 us; everything
// else is noise. S loads are non-temporal (won't fit in 192 MB L2, read once).
__global__ __launch_bounds__(256) void ap_rhs_wmma_kernel(
    const float* __restrict__ y,
    const float* __restrict__ S,
    const float* __restrict__ par,
    float* __restrict__ out)
{
    __shared__ float red[WAVES_PER_BLOCK * 16];

    const int tid  = threadIdx.x;
    const int wave = tid >> 5;      // 0..7
    const int lane = tid & 31;      // wave32
    const int half = lane >> 4;     // 0: K={0,1}, 1: K={2,3}
    const int l15  = lane & 15;     // row within tile (A) / column (B)

    const float* u = y;
    const float* v = y + DIM_D;

    const int row = blockIdx.x * 16 + l15;
    const int k0  = wave * K_PER_WAVE;

    // A: lane loads S[row][k + 2*half .. +1]  (8B per lane per WMMA)
    const float* aptr = S + (size_t)row * DIM_D + k0 + half * 2;
    // B: only column 0 is real; lanes 0/16 carry u[k + 2*half .. +1].
    // All lanes load (quarter-wave broadcast, cache hit) and multiply by a
    // per-lane mask -- arithmetic masking keeps the loop branch-free and
    // EXEC all-ones, as WMMA requires.
    const float* bptr = u + k0 + half * 2;

    const float mask  = (l15 == 0) ? 1.0f : 0.0f;
    const v2f   mask2 = {mask, mask};

    __builtin_prefetch(aptr, 0, 1);   // global_prefetch_b8 (CDNA5 path)

    v8f c0 = {};
    v8f c1 = {};

    #pragma unroll 2
    for (int k = 0; k < K_PER_WAVE; k += 8) {
        v2f a0  = __builtin_nontemporal_load((const v2f*)(aptr));
        v2f a1  = __builtin_nontemporal_load((const v2f*)(aptr + 4));
        v2f ub0 = *(const v2f*)(bptr);
        v2f ub1 = *(const v2f*)(bptr + 4);
        v2f b0  = ub0 * mask2;
        v2f b1  = ub1 * mask2;
        // (neg_a, A, neg_b, B, c_mod, C, reuse_a, reuse_b)
        c0 = __builtin_amdgcn_wmma_f32_16x16x4_f32(
                 false, a0, false, b0, (short)0, c0, false, false);
        c1 = __builtin_amdgcn_wmma_f32_16x16x4_f32(
                 false, a1, false, b1, (short)0, c1, false, false);
        aptr += 8;
        bptr += 8;
    }

    v8f c = c0 + c1;

    // Column 0 of C: lane 0 holds M=0..7, lane 16 holds M=8..15 (VGPRs 0..7).
    if (l15 == 0) {
        const int base = wave * 16 + half * 8;
        #pragma unroll
        for (int i = 0; i < 8; ++i) red[base + i] = c[i];
    }
    __syncthreads();

    if (tid < 16) {
        float mk = 0.0f;
        #pragma unroll
        for (int w = 0; w < WAVES_PER_BLOCK; ++w) mk += red[w * 16 + tid];

        const int   r  = blockIdx.x * 16 + tid;
        const float kk = 8.0f;
        const float ee = 0.01f;
        const float ur = u[r];
        const float vr = v[r];
        const float pr = par[r];

        const float pde1 = mk + kk * ur * (1.0f - ur) * (ur - pr) - ur * vr;
        const float pde2 = -ee * (kk * ur * (ur - pr - 1.0f) + vr);

        out[r]         = pde1;
        out[DIM_D + r] = pde2;
    }
}

extern "C" void kernel_launch(void* const* d_in, const int* in_sizes, int n_in,
                              void* d_out, int out_size, void* d_ws, size_t ws_size,
                              hipStream_t stream) {
    // setup_inputs order: t (scalar, unused), y [2D], S [D*D], par [D]
    const float* y   = (const float*)d_in[1];
    const float* S   = (const float*)d_in[2];
    const float* par = (const float*)d_in[3];
    float*       out = (float*)d_out;

    dim3 grid(DIM_D / 16);   // 512 row tiles
    dim3 block(256);         // 8 wave32s
    hipLaunchKernelGGL(ap_rhs_wmma_kernel, grid, block, 0, stream,
                       y, S, par, out);
}